// CompLayer_60395830116400
// MI455X (gfx1250) — compile-verified
//
#include <hip/hip_runtime.h>
#include <math.h>

#define HD 128          // hidden dim
#define TPB 256

typedef __attribute__((ext_vector_type(2))) float v2f;
typedef __attribute__((ext_vector_type(8))) float v8f;

// ---------------------------------------------------------------- utilities
__global__ void __launch_bounds__(TPB) zero_kernel(float* __restrict__ p, size_t n) {
    size_t i = (size_t)blockIdx.x * TPB + threadIdx.x;
    if (i < n) p[i] = 0.0f;
}

// order-preserving encode: float ordering == unsigned ordering, 0 == -inf sentinel
__device__ __forceinline__ unsigned enc_f32(float f) {
    unsigned b = __float_as_uint(f);
    return (b & 0x80000000u) ? ~b : (b | 0x80000000u);
}
__device__ __forceinline__ float dec_f32(unsigned e) {
    unsigned b = (e & 0x80000000u) ? (e & 0x7FFFFFFFu) : ~e;
    return __uint_as_float(b);
}

// ------------------------------------------- pass 1: edge score + segment max
// one wave (32 lanes) per edge; 4 floats per lane via float4
__global__ void __launch_bounds__(TPB) edge_score_kernel(
        const float* __restrict__ ent, const float* __restrict__ rel,
        const int* __restrict__ src, const int* __restrict__ dst,
        const int* __restrict__ rid,
        float* __restrict__ score, unsigned* __restrict__ segmax, int n_edges) {
    int e = blockIdx.x * (TPB / 32) + (threadIdx.x >> 5);
    if (e >= n_edges) return;
    int lane = threadIdx.x & 31;
    int s = src[e], d = dst[e], r = rid[e];
    const float4* es = (const float4*)(ent + (size_t)s * HD);
    const float4* ed = (const float4*)(ent + (size_t)d * HD);
    const float4* rr = (const float4*)(rel + (size_t)r * HD);
    float4 a = es[lane], b = rr[lane], c = ed[lane];
    float v = a.x*b.x*c.x + a.y*b.y*c.y + a.z*b.z*c.z + a.w*b.w*c.w;
    #pragma unroll
    for (int off = 16; off > 0; off >>= 1) v += __shfl_xor(v, off, 32);
    if (lane == 0) {
        score[e] = v;
        atomicMax(segmax + d, enc_f32(v));
    }
}

// ------------------------------------------- pass 2: exp(score - max) + denom
__global__ void __launch_bounds__(TPB) edge_exp_kernel(
        const int* __restrict__ dst, const unsigned* __restrict__ segmax,
        float* __restrict__ score, float* __restrict__ denom, int n_edges) {
    int e = blockIdx.x * TPB + threadIdx.x;
    if (e >= n_edges) return;
    int d = dst[e];
    float ex = __expf(score[e] - dec_f32(segmax[d]));
    score[e] = ex;                 // reuse buffer: score now holds numerator
    atomicAdd(denom + d, ex);
}

// ---------------------------- pass 3: alpha-weighted scatter-add of comp to dst
__global__ void __launch_bounds__(TPB) edge_aggr_kernel(
        const float* __restrict__ ent, const float* __restrict__ rel,
        const int* __restrict__ src, const int* __restrict__ dst,
        const int* __restrict__ rid,
        const float* __restrict__ score, const float* __restrict__ denom,
        float* __restrict__ neigh, int n_edges) {
    int e = blockIdx.x * (TPB / 32) + (threadIdx.x >> 5);
    if (e >= n_edges) return;
    int lane = threadIdx.x & 31;
    int s = src[e], d = dst[e], r = rid[e];
    float alpha = score[e] / denom[d];
    const float4* es = (const float4*)(ent + (size_t)s * HD);
    const float4* rr = (const float4*)(rel + (size_t)r * HD);
    float4 a = es[lane], b = rr[lane];
    float* out = neigh + (size_t)d * HD + lane * 4;
    atomicAdd(out + 0, a.x * b.x * alpha);
    atomicAdd(out + 1, a.y * b.y * alpha);
    atomicAdd(out + 2, a.z * b.z * alpha);
    atomicAdd(out + 3, a.w * b.w * alpha);
}

// ------------------------------- GEMM: H = neigh @ W via V_WMMA_F32_16X16X4_F32
// 8 waves per block, one 16x16 output tile per wave (N = 8 tiles of 16).
// grid.x = n_rows/16 M-tiles (50000/16 = 3125, exact -> EXEC all-ones everywhere).
__global__ void __launch_bounds__(TPB) gemm_wmma_kernel(
        const float* __restrict__ A,   // [n_rows, 128] neigh
        const float* __restrict__ W,   // [128, 128]
        float* __restrict__ H,         // [n_rows, 128]
        int n_rows) {
    const int wave = threadIdx.x >> 5;        // 0..7 -> N tile
    const int lane = threadIdx.x & 31;
    const int half = lane >> 4;               // 0 | 1
    const int l16  = lane & 15;
    const int m0 = blockIdx.x * 16;
    const int n0 = wave * 16;

    v8f c = {};
    const float* arow = A + (size_t)(m0 + l16) * HD;   // A: lane l16 holds row M=l16
    const float* wcol = W + n0 + l16;                  // B: lane l16 holds col N=l16

    #pragma unroll 4
    for (int k0 = 0; k0 < HD; k0 += 4) {
        int ka = k0 + 2 * half;                        // VGPRv -> K = v + 2*half
        v2f a, b;
        a.x = arow[ka];
        a.y = arow[ka + 1];
        b.x = wcol[(size_t)ka * HD];
        b.y = wcol[(size_t)(ka + 1) * HD];
        c = __builtin_amdgcn_wmma_f32_16x16x4_f32(
                /*neg_a=*/false, a, /*neg_b=*/false, b,
                /*c_mod=*/(short)0, c, /*reuse_a=*/false, /*reuse_b=*/false);
    }

    // D layout: VGPRr -> M = r + 8*half, N = l16
    float* hb = H + (size_t)(m0 + 8 * half) * HD + n0 + l16;
    #pragma unroll
    for (int r2 = 0; r2 < 8; ++r2) hb[(size_t)r2 * HD] = c[r2];
}

// --------------------------------------- BN column stats (sum, sum of squares)
// thread t of a block covers column t&127, row offset t>>7; grid-stride over rows
__global__ void __launch_bounds__(TPB) bn_stats_kernel(
        const float* __restrict__ H, float* __restrict__ colsum,
        float* __restrict__ colsumsq, int n_rows) {
    int col  = threadIdx.x & (HD - 1);
    int roff = threadIdx.x >> 7;               // 0..1
    float s = 0.0f, sq = 0.0f;
    for (int r = blockIdx.x * 2 + roff; r < n_rows; r += gridDim.x * 2) {
        float v = H[(size_t)r * HD + col];
        s += v; sq += v * v;
    }
    atomicAdd(colsum + col, s);
    atomicAdd(colsumsq + col, sq);
}

// ------------------------------------------------ BN normalize + tanh epilogue
__global__ void __launch_bounds__(TPB) bn_tanh_kernel(
        float* __restrict__ H, const float* __restrict__ colsum,
        const float* __restrict__ colsumsq, const float* __restrict__ gamma,
        const float* __restrict__ beta, int n_rows) {
    size_t i = (size_t)blockIdx.x * TPB + threadIdx.x;
    size_t total = (size_t)n_rows * HD;
    if (i >= total) return;
    int c = (int)(i & (HD - 1));
    float invN = 1.0f / (float)n_rows;
    float mu   = colsum[c] * invN;
    float var  = colsumsq[c] * invN - mu * mu;
    float istd = rsqrtf(var + 1e-5f);
    float v = (H[i] - mu) * istd * gamma[c] + beta[c];
    H[i] = tanhf(v);
}

extern "C" void kernel_launch(void* const* d_in, const int* in_sizes, int n_in,
                              void* d_out, int out_size, void* d_ws, size_t ws_size,
                              hipStream_t stream) {
    const float* ent   = (const float*)d_in[0];
    const float* rel   = (const float*)d_in[1];
    const float* W     = (const float*)d_in[2];
    const float* gamma = (const float*)d_in[3];
    const float* beta  = (const float*)d_in[4];
    const int*   src   = (const int*)d_in[5];
    const int*   dst   = (const int*)d_in[6];
    const int*   rid   = (const int*)d_in[7];
    float* H = (float*)d_out;

    const int N = in_sizes[0] / HD;   // 50000
    const int E = in_sizes[5];        // 800000

    // workspace layout (floats)
    float*    ws       = (float*)d_ws;
    float*    neigh    = ws;                               // N*128
    float*    score    = neigh + (size_t)N * HD;           // E
    unsigned* segmax   = (unsigned*)(score + E);           // N  (encoded, 0 == -inf)
    float*    denom    = (float*)(segmax + N);             // N
    float*    colsum   = denom + N;                        // 128
    float*    colsumsq = colsum + HD;                      // 128

    // zero the accumulated regions (neigh) and (segmax|denom|colsum|colsumsq)
    size_t zn = (size_t)N * HD;
    zero_kernel<<<(unsigned)((zn + TPB - 1) / TPB), TPB, 0, stream>>>(neigh, zn);
    size_t zt = (size_t)2 * N + 2 * HD;
    zero_kernel<<<(unsigned)((zt + TPB - 1) / TPB), TPB, 0, stream>>>((float*)segmax, zt);

    const int edges_per_blk = TPB / 32;
    edge_score_kernel<<<(unsigned)((E + edges_per_blk - 1) / edges_per_blk), TPB, 0, stream>>>(
        ent, rel, src, dst, rid, score, segmax, E);
    edge_exp_kernel<<<(unsigned)((E + TPB - 1) / TPB), TPB, 0, stream>>>(
        dst, segmax, score, denom, E);
    edge_aggr_kernel<<<(unsigned)((E + edges_per_blk - 1) / edges_per_blk), TPB, 0, stream>>>(
        ent, rel, src, dst, rid, score, denom, neigh, E);

    gemm_wmma_kernel<<<(unsigned)(N / 16), TPB, 0, stream>>>(neigh, W, H, N);

    bn_stats_kernel<<<256, TPB, 0, stream>>>(H, colsum, colsumsq, N);
    bn_tanh_kernel<<<(unsigned)(((size_t)N * HD + TPB - 1) / TPB), TPB, 0, stream>>>(
        H, colsum, colsumsq, gamma, beta, N);
}